// HolographicAssociativeMemory_22643067585265
// MI455X (gfx1250) — compile-verified
//
#include <hip/hip_runtime.h>

// ---------------------------------------------------------------------------
// HolographicAssociativeMemory on MI455X (gfx1250)
//   stage 1: QR = F128 @ q              (complex = DFT-matrix x real)
//   stage 2: QF = QR @ F256             (complex x complex; F256 symmetric ->
//                                        B fragments read transposed = b64 loads)
//   stage 3: per m,w:  rec = Finv @ (QF_row * H * beam); out = mean|rec| thresholded
// All GEMM stages run on V_WMMA_F32_16X16X4_F32 (fp32 needed for DFT accuracy).
// ---------------------------------------------------------------------------

typedef __attribute__((ext_vector_type(2))) float v2f;
typedef __attribute__((ext_vector_type(8))) float v8f;

#define BATCH 32
#define RDIM  128
#define PD    256
#define MEM   512
#define NW    3
#define PI_F  3.14159265358979323846f
#define XPAD  4   // LDS row skew: lanes 0-15 hit banks 4n+k, lanes 16-31 4n+2+k

// workspace layout (float offsets)
constexpr int OFF_F128R = 0;
constexpr int OFF_F128I = OFF_F128R + RDIM * RDIM;
constexpr int OFF_F256R = OFF_F128I + RDIM * RDIM;
constexpr int OFF_F256I = OFF_F256R + PD * PD;
constexpr int OFF_FINVR = OFF_F256I + PD * PD;
constexpr int OFF_FINVI = OFF_FINVR + PD * PD;
constexpr int OFF_QRR   = OFF_FINVI + PD * PD;
constexpr int OFF_QRI   = OFF_QRR + BATCH * RDIM * PD;
constexpr int OFF_QFR   = OFF_QRI + BATCH * RDIM * PD;
constexpr int OFF_QFI   = OFF_QFR + BATCH * RDIM * PD;

__device__ __forceinline__ v8f wmma_f32(v2f a, v2f b, v8f c) {
  // 8 args: (neg_a, A, neg_b, B, c_mod, C, reuse_a, reuse_b)
  // f32 WMMA: NEG only allows C-negate (ISA 7.12), so complex-GEMM subtraction
  // is done in VALU at store time.
  return __builtin_amdgcn_wmma_f32_16x16x4_f32(false, a, false, b, (short)0, c,
                                               false, false);
}

__device__ __forceinline__ v2f ld2(const float* p) { return *(const v2f*)p; }

// ---------------------------------------------------------------------------
// Stage 0: DFT twiddle matrices. Integer-mod phase keeps fp32 angles exact.
// ---------------------------------------------------------------------------
__global__ __launch_bounds__(256) void twiddle_kernel(float* __restrict__ ws) {
  int idx = blockIdx.x * blockDim.x + threadIdx.x;  // 0 .. 65535
  if (idx < RDIM * RDIM) {
    int r = idx >> 7, c = idx & (RDIM - 1);
    int t = (r * c) & (RDIM - 1);
    float ang = -2.0f * PI_F * (float)t / (float)RDIM;
    float s, co;
    __sincosf(ang, &s, &co);
    ws[OFF_F128R + idx] = co;
    ws[OFF_F128I + idx] = s;
  }
  if (idx < PD * PD) {
    int r = idx >> 8, c = idx & (PD - 1);
    int t = (r * c) & (PD - 1);
    float ang = -2.0f * PI_F * (float)t / (float)PD;
    float s, co;
    __sincosf(ang, &s, &co);
    ws[OFF_F256R + idx] = co;
    ws[OFF_F256I + idx] = s;
    // Finv = conj(F256) / 256   (inverse DFT)
    ws[OFF_FINVR + idx] = co * (1.0f / (float)PD);
    ws[OFF_FINVI + idx] = -s * (1.0f / (float)PD);
  }
}

// ---------------------------------------------------------------------------
// Stage 1: QR[b] (128x256 complex) = F128 (128x128 complex) @ q[b] (128x256 real)
// One wave per 16x16 tile; K stepped by 4 (V_WMMA_F32_16X16X4_F32).
// A-fragment layout: lane m=lane&15; VGPR pair holds K = k0 + 2*(lane>=16) {+0,+1}.
// ---------------------------------------------------------------------------
__global__ __launch_bounds__(256) void rowfft_kernel(const float* __restrict__ stim,
                                                     float* __restrict__ ws) {
  const float* F128r = ws + OFF_F128R;
  const float* F128i = ws + OFF_F128I;
  float* QRr = ws + OFF_QRR;
  float* QRi = ws + OFF_QRI;

  int b    = blockIdx.y;
  int wv   = threadIdx.x >> 5;
  int lane = threadIdx.x & 31;
  int tile = blockIdx.x * 8 + wv;  // 128 tiles: 8 Mtiles x 16 Ntiles
  int Mtile = tile >> 4;
  int Ntile = tile & 15;

  int am   = lane & 15;
  int hk   = (lane >> 4) << 1;  // 0 or 2
  int arow = (Mtile * 16 + am) * RDIM;
  int bcol = Ntile * 16 + am;
  const float* qb = stim + b * (RDIM * PD);

  v8f accR = {};
  v8f accI = {};
  for (int k0 = 0; k0 < RDIM; k0 += 4) {
    int ka = k0 + hk;
    v2f ar = ld2(&F128r[arow + ka]);                  // global_load_b64
    v2f ai = ld2(&F128i[arow + ka]);
    v2f bb = {qb[ka * PD + bcol], qb[(ka + 1) * PD + bcol]};  // q not symmetric
    accR = wmma_f32(ar, bb, accR);
    accI = wmma_f32(ai, bb, accI);
  }

  int rbase = Mtile * 16 + ((lane >> 4) << 3);
  int col   = Ntile * 16 + am;
  float* oR = QRr + b * (RDIM * PD);
  float* oI = QRi + b * (RDIM * PD);
#pragma unroll
  for (int i = 0; i < 8; ++i) {
    oR[(rbase + i) * PD + col] = accR[i];
    oI[(rbase + i) * PD + col] = accI[i];
  }
}

// ---------------------------------------------------------------------------
// Stage 2: QF[b] = QR[b] (128x256 complex) @ F256 (256x256 complex)
// F256 is symmetric, so B fragments {F[k][n], F[k+1][n]} are read as
// {F[n][k], F[n][k+1]} -> contiguous global_load_b64.
// ---------------------------------------------------------------------------
__global__ __launch_bounds__(256) void colfft_kernel(float* __restrict__ ws) {
  const float* F256r = ws + OFF_F256R;
  const float* F256i = ws + OFF_F256I;
  const float* QRr   = ws + OFF_QRR;
  const float* QRi   = ws + OFF_QRI;
  float* QFr = ws + OFF_QFR;
  float* QFi = ws + OFF_QFI;

  int b    = blockIdx.y;
  int wv   = threadIdx.x >> 5;
  int lane = threadIdx.x & 31;
  int tile = blockIdx.x * 8 + wv;  // 8 Mtiles x 16 Ntiles
  int Mtile = tile >> 4;
  int Ntile = tile & 15;

  int am   = lane & 15;
  int hk   = (lane >> 4) << 1;
  int arow = (Mtile * 16 + am) * PD;
  int brow = (Ntile * 16 + am) * PD;  // transposed (symmetric) B indexing
  const float* Ar = QRr + b * (RDIM * PD);
  const float* Ai = QRi + b * (RDIM * PD);

  v8f cRR = {}, cII = {}, cRI = {}, cIR = {};
  for (int k0 = 0; k0 < PD; k0 += 4) {
    int ka = k0 + hk;
    v2f ar = ld2(&Ar[arow + ka]);
    v2f ai = ld2(&Ai[arow + ka]);
    v2f br = ld2(&F256r[brow + ka]);   // == {F256[ka][n], F256[ka+1][n]}
    v2f bi = ld2(&F256i[brow + ka]);
    cRR = wmma_f32(ar, br, cRR);
    cII = wmma_f32(ai, bi, cII);
    cRI = wmma_f32(ar, bi, cRI);
    cIR = wmma_f32(ai, br, cIR);
  }

  int rbase = Mtile * 16 + ((lane >> 4) << 3);
  int col   = Ntile * 16 + am;
  float* oR = QFr + b * (RDIM * PD);
  float* oI = QFi + b * (RDIM * PD);
#pragma unroll
  for (int i = 0; i < 8; ++i) {
    oR[(rbase + i) * PD + col] = cRR[i] - cII[i];
    oI[(rbase + i) * PD + col] = cRI[i] + cIR[i];
  }
}

// ---------------------------------------------------------------------------
// Stage 3: one block per memory slot m.
//   loop w: Xt[b][j] = QF[b, m%128, j] * H[m,j,w] * beam_w   (LDS, transposed:
//           k-pairs contiguous -> ds_load_b64; row skew 4 -> conflict-free)
//           rec (256x32 complex) = Finv @ X  via WMMA, Finv streamed from L2
//           colsum[b] += sum_j |rec[j][b]|   (ds_add_f32 shared atomics)
//   out[b, m] = threshold(colsum[b] / (256*3))
// 512 threads = 16 waves; wave wv owns M-tile wv for both 16-wide N tiles.
// ---------------------------------------------------------------------------
__global__ __launch_bounds__(512) void recon_kernel(const float* __restrict__ Hr,
                                                    const float* __restrict__ Hi,
                                                    const float* __restrict__ ws,
                                                    float* __restrict__ out) {
  __shared__ float Xtr[BATCH][PD + XPAD];
  __shared__ float Xti[BATCH][PD + XPAD];
  __shared__ float colsum[BATCH];

  const float* QFr   = ws + OFF_QFR;
  const float* QFi   = ws + OFF_QFI;
  const float* FINVr = ws + OFF_FINVR;
  const float* FINVi = ws + OFF_FINVI;

  int m   = blockIdx.x;       // 0..511
  int r   = m & (RDIM - 1);   // source FFT row
  int tid = threadIdx.x;
  int lane = tid & 31;
  int wv   = tid >> 5;        // 0..15 -> M-tile

  if (tid < BATCH) colsum[tid] = 0.0f;

  int am   = lane & 15;
  int hk   = (lane >> 4) << 1;
  int arow = (wv * 16 + am) * PD;
  __builtin_prefetch(FINVr + arow, 0, 3);

  // X-build assignment: thread -> one batch row, 16 consecutive-j strips
  int xb  = tid >> 4;   // 0..31
  int xj0 = tid & 15;   // j = xj0 + 16*it  (half-wave reads 16 consecutive j)
  const float* qfr = QFr + xb * (RDIM * PD) + r * PD;
  const float* qfi = QFi + xb * (RDIM * PD) + r * PD;

  for (int w = 0; w < NW; ++w) {
    __syncthreads();  // colsum zeroed / previous iteration's X reads finished

    // ---- build Xt[b][j] = QF * (H * beam), coalesced along j ----
    {
      float bs, bc;
      __sincosf(PI_F * (float)w, &bs, &bc);  // linspace(0, 2pi, 3)
#pragma unroll
      for (int it = 0; it < 16; ++it) {
        int j = xj0 + (it << 4);
        float hr = Hr[(m * PD + j) * NW + w];
        float hi = Hi[(m * PD + j) * NW + w];
        float fr = hr * bc - hi * bs;  // H * beam
        float fi = hr * bs + hi * bc;
        float qr = qfr[j];
        float qi = qfi[j];
        Xtr[xb][j] = qr * fr - qi * fi;
        Xti[xb][j] = qr * fi + qi * fr;
      }
    }
    __syncthreads();

    // ---- rec = Finv @ X : complex GEMM, 2 N-tiles per wave ----
    v8f cRR0 = {}, cII0 = {}, cRI0 = {}, cIR0 = {};
    v8f cRR1 = {}, cII1 = {}, cRI1 = {}, cIR1 = {};
    for (int k0 = 0; k0 < PD; k0 += 4) {
      int ka = k0 + hk;
      v2f ar  = ld2(&FINVr[arow + ka]);       // global_load_b64, L2-resident
      v2f ai  = ld2(&FINVi[arow + ka]);
      v2f br0 = ld2(&Xtr[am][ka]);            // ds_load_b64, conflict-free
      v2f bi0 = ld2(&Xti[am][ka]);
      v2f br1 = ld2(&Xtr[am + 16][ka]);
      v2f bi1 = ld2(&Xti[am + 16][ka]);
      cRR0 = wmma_f32(ar, br0, cRR0);
      cII0 = wmma_f32(ai, bi0, cII0);
      cRI0 = wmma_f32(ar, bi0, cRI0);
      cIR0 = wmma_f32(ai, br0, cIR0);
      cRR1 = wmma_f32(ar, br1, cRR1);
      cII1 = wmma_f32(ai, bi1, cII1);
      cRI1 = wmma_f32(ar, bi1, cRI1);
      cIR1 = wmma_f32(ai, br1, cIR1);
    }

    // ---- |rec| partial sums per output column (= batch index) ----
    float p0 = 0.0f, p1 = 0.0f;
#pragma unroll
    for (int i = 0; i < 8; ++i) {
      float rr = cRR0[i] - cII0[i];
      float ii = cRI0[i] + cIR0[i];
      p0 += sqrtf(rr * rr + ii * ii);
      rr = cRR1[i] - cII1[i];
      ii = cRI1[i] + cIR1[i];
      p1 += sqrtf(rr * rr + ii * ii);
    }
    atomicAdd(&colsum[am], p0);
    atomicAdd(&colsum[am + 16], p1);
  }

  __syncthreads();
  if (tid < BATCH) {
    float v = colsum[tid] * (1.0f / (float)(PD * NW));  // mean over pattern & wavelength
    out[tid * MEM + m] = (v > 0.3f) ? v : 0.0f;         // ATTENTION_FOCUS=1, threshold
  }
}

// ---------------------------------------------------------------------------
extern "C" void kernel_launch(void* const* d_in, const int* in_sizes, int n_in,
                              void* d_out, int out_size, void* d_ws, size_t ws_size,
                              hipStream_t stream) {
  const float* stim = (const float*)d_in[0];  // (32, 128*256) fp32
  const float* Hr   = (const float*)d_in[1];  // (512, 256, 3) fp32
  const float* Hi   = (const float*)d_in[2];  // (512, 256, 3) fp32
  float* out = (float*)d_out;                 // (32, 512) fp32
  float* ws  = (float*)d_ws;                  // ~17.1 MB of fp32 scratch

  twiddle_kernel<<<256, 256, 0, stream>>>(ws);
  rowfft_kernel<<<dim3(16, BATCH), 256, 0, stream>>>(stim, ws);
  colfft_kernel<<<dim3(16, BATCH), 256, 0, stream>>>(ws);
  recon_kernel<<<MEM, 512, 0, stream>>>(Hr, Hi, ws, out);
}